// GraphAttentionBlock_29429115912696
// MI455X (gfx1250) — compile-verified
//
#include <hip/hip_runtime.h>

// Graph attention block for MI455X (gfx1250, wave32, WMMA).
// Reference: Q/K/V = h@W + b ; score = (adj * QK^T + mask)/sqrt(DH);
// out = softmax(score) @ V, per head.

typedef float v2f __attribute__((ext_vector_type(2)));
typedef float v8f __attribute__((ext_vector_type(8)));

#define NTOK 4096
#define DM   512
#define NH   8
#define DHD  64
#define SCALE 0.125f   // 1/sqrt(64)

static __device__ __forceinline__ v8f wmma_f32(v2f a, v2f b, v8f c) {
  // D = A(16x4) * B(4x16) + C(16x16), full fp32
  return __builtin_amdgcn_wmma_f32_16x16x4_f32(
      /*neg_a=*/false, a, /*neg_b=*/false, b,
      /*c_mod=*/(short)0, c, /*reuse_a=*/false, /*reuse_b=*/false);
}

// ---------------------------------------------------------------------------
// Kernel 1: fused Q/K/V projection. One wave computes one 16x16 output tile.
// tiles = 3 mats * 256 row-tiles * 32 col-tiles = 24576; 8 waves / block.
// ---------------------------------------------------------------------------
__global__ __launch_bounds__(256) void qkv_proj_kernel(
    const float* __restrict__ h,
    const float* __restrict__ Wq, const float* __restrict__ bq,
    const float* __restrict__ Wk, const float* __restrict__ bk,
    const float* __restrict__ Wv, const float* __restrict__ bv,
    float* __restrict__ Qo, float* __restrict__ Ko, float* __restrict__ Vo)
{
  const int lane = threadIdx.x & 31;
  const int wid  = threadIdx.x >> 5;
  const int tile = blockIdx.x * 8 + wid;
  const int mat  = tile >> 13;           // / (256*32)
  const int rem  = tile & 8191;
  const int rowT = rem >> 5;
  const int colT = rem & 31;

  const float* W; const float* bias; float* Out;
  if (mat == 0)      { W = Wq; bias = bq; Out = Qo; }
  else if (mat == 1) { W = Wk; bias = bk; Out = Ko; }
  else               { W = Wv; bias = bv; Out = Vo; }

  const int mr   = lane & 15;            // A-layout: lane%16 = M row
  const int half = lane >> 4;            // lane half selects K pair
  const int n    = colT * 16 + mr;       // B/C-layout: lane%16 = N col
  const float* hrow = h + (size_t)(rowT * 16 + mr) * DM;

  v8f c = {};
  for (int k0 = 0; k0 < DM; k0 += 4) {
    const int ka = k0 + 2 * half;
    v2f a = *(const v2f*)(hrow + ka);    // A: K = {ka, ka+1}
    v2f b;
    b.x = W[(size_t)(ka + 0) * DM + n];  // B: rows K = ka, ka+1 at column n
    b.y = W[(size_t)(ka + 1) * DM + n];
    c = wmma_f32(a, b, c);
  }
  const float bn = bias[n];
  #pragma unroll
  for (int v = 0; v < 8; ++v)
    Out[(size_t)(rowT * 16 + v + 8 * half) * DM + n] = c[v] + bn;
}

// ---------------------------------------------------------------------------
// Kernel 2: flash-attention over adjacency-weighted scores.
// One wave handles one (head, 16-row block); 4 independent waves per block.
// ---------------------------------------------------------------------------
__global__ __launch_bounds__(128) void graph_attn_kernel(
    const float* __restrict__ Q, const float* __restrict__ K,
    const float* __restrict__ V,
    const float* __restrict__ adj, const float* __restrict__ msk,
    float* __restrict__ out)
{
  // P-transpose staging: 16x16 tile, row stride 18 dwords:
  //  - rows r and r+8 land on disjoint bank groups (18*8 % 64 == 16)
  //  - 18*m + 4*s + 2*half is even -> 8B-aligned ds_load_b64
  __shared__ float pl[4][16 * 18];

  const int lane = threadIdx.x & 31;
  const int wid  = threadIdx.x >> 5;
  const int task = blockIdx.x * 4 + wid;     // 0 .. 2047
  const int head = task >> 8;
  const int rowB = (task & 255) * 16;
  const int nn   = lane & 15;
  const int half = lane >> 4;
  float* plw = pl[wid];

  // Preload Q A-fragments for the whole DH=64 reduction (16 WMMA steps).
  const float* qrow = Q + (size_t)(rowB + nn) * DM + head * DHD;
  v2f qa[16];
  #pragma unroll
  for (int s = 0; s < 16; ++s)
    qa[s] = *(const v2f*)(qrow + 4 * s + 2 * half);

  v8f acc[4] = {};                    // O tiles: dh columns 16t .. 16t+15
  float mrow[8], lrow[8];
  #pragma unroll
  for (int v = 0; v < 8; ++v) { mrow[v] = -3.0e38f; lrow[v] = 0.0f; }

  for (int j0 = 0; j0 < NTOK; j0 += 16) {
    // ---- S = Q @ K^T for 16 columns (K-dim = 64 -> 16 f32 WMMAs) ----
    v8f sf = {};
    const float* krow = K + (size_t)(j0 + nn) * DM + head * DHD;
    #pragma unroll
    for (int s = 0; s < 16; ++s) {
      v2f b = *(const v2f*)(krow + 4 * s + 2 * half);  // B[k][n] = K[j0+n][k]
      sf = wmma_f32(qa[s], b, sf);
    }

    if (j0 + 16 < NTOK)  // stream-ahead hint for the adj tile we touch next
      __builtin_prefetch(&adj[(size_t)(rowB + 8 * half) * NTOK + j0 + 16 + nn], 0, 1);

    // ---- adjacency weighting + additive mask + online softmax ----
    float p[8];
    #pragma unroll
    for (int v = 0; v < 8; ++v) {
      const size_t e = (size_t)(rowB + v + 8 * half) * NTOK + j0 + nn;
      const float sc = (adj[e] * sf[v] + msk[e]) * SCALE;
      float bm = sc;                               // row-max across N lanes
      bm = fmaxf(bm, __shfl_xor(bm, 1, 16));
      bm = fmaxf(bm, __shfl_xor(bm, 2, 16));
      bm = fmaxf(bm, __shfl_xor(bm, 4, 16));
      bm = fmaxf(bm, __shfl_xor(bm, 8, 16));
      const float newm = fmaxf(mrow[v], bm);
      const float f    = __expf(mrow[v] - newm);   // first block: exp(-inf)=0
      const float pv   = __expf(sc - newm);
      float bs = pv;                               // row-sum across N lanes
      bs += __shfl_xor(bs, 1, 16);
      bs += __shfl_xor(bs, 2, 16);
      bs += __shfl_xor(bs, 4, 16);
      bs += __shfl_xor(bs, 8, 16);
      lrow[v] = lrow[v] * f + bs;
      mrow[v] = newm;
      p[v] = pv;
      #pragma unroll
      for (int t = 0; t < 4; ++t) acc[t][v] *= f;  // rescale accumulator
    }

    // ---- transpose P (C layout -> A layout) through LDS ----
    #pragma unroll
    for (int v = 0; v < 8; ++v)
      plw[(v + 8 * half) * 18 + nn] = p[v];
    __builtin_amdgcn_wave_barrier();   // same-wave LDS ops are in-order

    // ---- O += P @ V  (K-dim = 16 -> 4 steps x 4 dh tiles) ----
    #pragma unroll
    for (int s = 0; s < 4; ++s) {
      v2f a = *(const v2f*)&plw[nn * 18 + 4 * s + 2 * half];
      const float* v0 = V + (size_t)(j0 + 4 * s + 2 * half) * DM + head * DHD;
      #pragma unroll
      for (int t = 0; t < 4; ++t) {
        v2f b;
        b.x = v0[t * 16 + nn];         // V[j][dh], consecutive j rows
        b.y = v0[DM + t * 16 + nn];
        acc[t] = wmma_f32(a, b, acc[t]);
      }
    }
  }

  // ---- normalize and store ----
  #pragma unroll
  for (int t = 0; t < 4; ++t) {
    #pragma unroll
    for (int v = 0; v < 8; ++v) {
      out[(size_t)(rowB + v + 8 * half) * DM + head * DHD + t * 16 + nn] =
          acc[t][v] / lrow[v];
    }
  }
}

// ---------------------------------------------------------------------------
extern "C" void kernel_launch(void* const* d_in, const int* in_sizes, int n_in,
                              void* d_out, int out_size, void* d_ws, size_t ws_size,
                              hipStream_t stream) {
  (void)in_sizes; (void)n_in; (void)out_size; (void)ws_size;
  const float* h   = (const float*)d_in[0];
  const float* adj = (const float*)d_in[1];
  const float* msk = (const float*)d_in[2];
  const float* Wq  = (const float*)d_in[3];
  const float* bq  = (const float*)d_in[4];
  const float* Wk  = (const float*)d_in[5];
  const float* bk  = (const float*)d_in[6];
  const float* Wv  = (const float*)d_in[7];
  const float* bv  = (const float*)d_in[8];
  float* out = (float*)d_out;

  float* Qo = (float*)d_ws;                   // 3 x (4096*512) f32 = 24 MB
  float* Ko = Qo + (size_t)NTOK * DM;
  float* Vo = Ko + (size_t)NTOK * DM;

  // 24576 tiles / 8 waves per block
  qkv_proj_kernel<<<3 * 256 * 32 / 8, 256, 0, stream>>>(
      h, Wq, bq, Wk, bk, Wv, bv, Qo, Ko, Vo);

  // 2048 (head, row-block) tasks / 4 waves per block
  graph_attn_kernel<<<(NH * 256) / 4, 128, 0, stream>>>(
      Qo, Ko, Vo, adj, msk, out);
}